// MaxDistance_10625749090596
// MI455X (gfx1250) — compile-verified
//
#include <hip/hip_runtime.h>
#include <hip/hip_bf16.h>
#include <math.h>

// MaxDistance (Hausdorff-style) on MI455X / gfx1250.
// Uses V_WMMA_F32_16X16X4_F32: A row n = [-2x0,-2x1,-2x2,1], B col m = [y0,y1,y2,||y||^2]
// => C[n,m] = ||y_m||^2 - 2 x.y ; d2 = C + ||x_n||^2 (added after min over m).

typedef __attribute__((ext_vector_type(2))) float v2f;
typedef __attribute__((ext_vector_type(8))) float v8f;

#define BATCHES   8
#define NPOINTS   4096
#define MTILES    (NPOINTS / 16)   // 256 column tiles
#define NTILES    (NPOINTS / 16)   // 256 row tiles per batch
#define WAVES_TOT (BATCHES * NTILES)

// One wave handles one (batch, 16-row tile) and sweeps all 4096 columns.
__global__ __launch_bounds__(256) void hausdorff_dir_kernel(
    const float* __restrict__ P,   // "A-side" points [B, N, 3]
    const float* __restrict__ Q,   // "B-side" points [B, N, 3]
    int* __restrict__ dirmax)      // [B] float-bits, atomicMax (values >= 0)
{
    const int lane   = threadIdx.x & 31;
    const int waveId = blockIdx.x * (blockDim.x >> 5) + (threadIdx.x >> 5);
    const int b      = waveId >> 8;        // / NTILES
    const int tile   = waveId & (NTILES - 1);
    const int n0     = tile * 16;

    const float* __restrict__ Pb = P + (size_t)b * NPOINTS * 3;
    const float* __restrict__ Qb = Q + (size_t)b * NPOINTS * 3;

    // ---- Build A (16x4 f32): lanes 0-15 hold K=0,1 ; lanes 16-31 hold K=2,3 ; M = lane%16
    const int  mrow = lane & 15;
    const bool hi   = (lane >= 16);
    const float* pr = Pb + (size_t)(n0 + mrow) * 3;
    const float px0 = pr[0], px1 = pr[1], px2 = pr[2];
    v2f A;
    A.x = hi ? (-2.0f * px2) : (-2.0f * px0);
    A.y = hi ? 1.0f          : (-2.0f * px1);

    v8f runMin;
#pragma unroll
    for (int r = 0; r < 8; ++r) runMin[r] = 3.0e38f;

    const v8f Czero = {};

    // ---- Sweep all column tiles
    for (int mt = 0; mt < MTILES; ++mt) {
        const int col = mt * 16 + (lane & 15);
        const float* qc = Qb + (size_t)col * 3;
        const float q0 = qc[0], q1 = qc[1], q2 = qc[2];
        const float qn = q0 * q0 + q1 * q1 + q2 * q2;
        // B (4x16 f32): lanes 0-15 hold K=0,1 ; lanes 16-31 hold K=2,3 ; N = lane%16
        v2f Bv;
        Bv.x = hi ? q2 : q0;
        Bv.y = hi ? qn : q1;

        // C[n,m] = ||y_m||^2 - 2 x_n . y_m   (full fp32 on the matrix pipe)
        v8f D = __builtin_amdgcn_wmma_f32_16x16x4_f32(
            /*neg_a=*/false, A, /*neg_b=*/false, Bv,
            /*c_mod=*/(short)0, Czero, /*reuse_a=*/false, /*reuse_b=*/false);

#pragma unroll
        for (int r = 0; r < 8; ++r) runMin[r] = fminf(runMin[r], D[r]);
    }

    // ---- Min over N (columns): N is striped across lanes within each 16-lane half.
#pragma unroll
    for (int mask = 1; mask <= 8; mask <<= 1) {
#pragma unroll
        for (int r = 0; r < 8; ++r)
            runMin[r] = fminf(runMin[r], __shfl_xor(runMin[r], mask, 32));
    }

    // ---- Redistribute: lane t (t = lane&15) picks row t.
    // Row t lives in register (t&7) of half (t>>3); every lane of that half has it.
    const int t = lane & 15;
    float v = runMin[0];
#pragma unroll
    for (int r = 1; r < 8; ++r) v = ((t & 7) == r) ? runMin[r] : v;
    const int srcLane = (t & 7) | ((t >> 3) << 4);
    v = __shfl(v, srcLane, 32);

    // ---- Add ||x_row||^2, clamp at 0 (monotone => exact vs clamping before min)
    const float* pt = Pb + (size_t)(n0 + t) * 3;
    const float p0 = pt[0], p1 = pt[1], p2c = pt[2];
    float d2 = fmaxf(v + (p0 * p0 + p1 * p1 + p2c * p2c), 0.0f);

    // ---- Max over the 16 rows of this tile (upper half holds duplicates; harmless)
#pragma unroll
    for (int mask = 1; mask <= 16; mask <<= 1)
        d2 = fmaxf(d2, __shfl_xor(d2, mask, 32));

    if (lane == 0)
        atomicMax(&dirmax[b], __float_as_int(d2));  // d2 >= 0 => int-compare == float-compare
}

__global__ void hausdorff_init_kernel(int* __restrict__ ws) {
    const int i = threadIdx.x;
    if (i < 2 * BATCHES) ws[i] = 0;  // bits of 0.0f
}

__global__ void hausdorff_finalize_kernel(const int* __restrict__ ws,
                                          float* __restrict__ out) {
    if (threadIdx.x == 0 && blockIdx.x == 0) {
        float acc = 0.0f;
#pragma unroll
        for (int b = 0; b < BATCHES; ++b) {
            const float dxy = __int_as_float(ws[b]);
            const float dyx = __int_as_float(ws[BATCHES + b]);
            acc += sqrtf(fmaxf(dxy, dyx));
        }
        out[0] = acc * (1.0f / BATCHES);
    }
}

extern "C" void kernel_launch(void* const* d_in, const int* in_sizes, int n_in,
                              void* d_out, int out_size, void* d_ws, size_t ws_size,
                              hipStream_t stream) {
    (void)in_sizes; (void)n_in; (void)out_size; (void)ws_size;
    const float* x = (const float*)d_in[0];
    const float* y = (const float*)d_in[1];
    float* out = (float*)d_out;
    int*   ws  = (int*)d_ws;   // ws[0..7] = directed x->y, ws[8..15] = directed y->x

    hipLaunchKernelGGL(hausdorff_init_kernel, dim3(1), dim3(32), 0, stream, ws);

    const int wavesPerBlock = 256 / 32;               // 8
    const int blocks = WAVES_TOT / wavesPerBlock;     // 2048 waves -> 256 blocks
    hipLaunchKernelGGL(hausdorff_dir_kernel, dim3(blocks), dim3(256), 0, stream,
                       x, y, ws);                     // x -> y
    hipLaunchKernelGGL(hausdorff_dir_kernel, dim3(blocks), dim3(256), 0, stream,
                       y, x, ws + BATCHES);           // y -> x

    hipLaunchKernelGGL(hausdorff_finalize_kernel, dim3(1), dim3(32), 0, stream, ws, out);
}